// BaseSmear_64046552318459
// MI455X (gfx1250) — compile-verified
//
#include <hip/hip_runtime.h>

typedef __attribute__((ext_vector_type(2))) float v2f;
typedef __attribute__((ext_vector_type(8))) float v8f;

#define NIMG  8
#define CCH   32
#define HH    480
#define WW    480
#define NPTS  (64 * 64 * 64)
#define OCH   37          // 32 feats + depth + validity + 3 view_dir
#define HW    ((size_t)HH * (size_t)WW)

__global__ __launch_bounds__(256) void proj_gather_wmma_kernel(
    const float* __restrict__ coords,   // (3, N)
    const float* __restrict__ images,   // (I, C, H, W)
    const float* __restrict__ trans,    // (I, 3, 4)
    const float* __restrict__ tcw,      // (I, 4, 4)
    float* __restrict__ out)            // (I, 37, N)
{
    // ---- per-block: camera centers cam[i][l] = -(R^T t)[l] into LDS ----
    __shared__ float s_cam[NIMG][3];
    if (threadIdx.x < NIMG * 3) {
        const int img = threadIdx.x / 3;
        const int l   = threadIdx.x % 3;
        const float* T = tcw + img * 16;
        // R[k][l] = T[k*4+l], t[k] = T[k*4+3]
        s_cam[img][l] = -(T[l] * T[3] + T[4 + l] * T[7] + T[8 + l] * T[11]);
    }
    __syncthreads();

    const int lane  = threadIdx.x & 31;
    const int wib   = threadIdx.x >> 5;                 // wave in block
    const int group = blockIdx.x * 8 + wib;             // 16-point group
    const int p     = lane & 15;                        // point (column) for this lane
    const int n     = group * 16 + p;                   // global point index
    const bool hi   = lane >= 16;

    // ---- point coordinates (both half-waves load same 16 points) ----
    const float px = coords[0 * NPTS + n];
    const float py = coords[1 * NPTS + n];
    const float pz = coords[2 * NPTS + n];

    // ---- B operand: 4x16, column p = (x, y, z, 1)
    //      lanes 0-15 hold K=0,1 ; lanes 16-31 hold K=2,3
    v2f b;
    b.x = hi ? pz : px;
    b.y = hi ? 1.0f : py;

    // ---- A operands: row m = 4*g + r;  r<3 -> trans[g][r][:], r==3 -> T_cw[g][2][:]
    //      WMMA0 covers images 0..3, WMMA1 covers images 4..7
    const int r  = p & 3;
    const int k0 = hi ? 2 : 0;
    v2f a0, a1;
    {
        const int g0 = p >> 2;
        const int g1 = g0 + 4;
        const float* row0 = (r < 3) ? (trans + ((size_t)g0 * 3 + r) * 4)
                                    : (tcw + (size_t)g0 * 16 + 8);
        const float* row1 = (r < 3) ? (trans + ((size_t)g1 * 3 + r) * 4)
                                    : (tcw + (size_t)g1 * 16 + 8);
        a0.x = row0[k0]; a0.y = row0[k0 + 1];
        a1.x = row1[k0]; a1.y = row1[k0 + 1];
    }

    // ---- projections + depth for 16 points x 8 images via two WMMAs ----
    v8f cz = {};
    // args: (neg_a, A, neg_b, B, c_mod, C, reuse_a, reuse_b)
    v8f d0 = __builtin_amdgcn_wmma_f32_16x16x4_f32(false, a0, false, b,
                                                   (short)0, cz, false, false);
    v8f d1 = __builtin_amdgcn_wmma_f32_16x16x4_f32(false, a1, false, b,
                                                   (short)0, cz, false, false);

    // lane holds D rows (hi?8:0)+0..7  ->  images (hi?2:0)+{0,1} (+4 for d1),
    // each image = 4 consecutive rows: x, y, z, depth
    const int ib = hi ? 2 : 0;

    auto process = [&](int img, float X, float Yv, float Zv, float depth) {
        const float zsafe = (fabsf(Zv) < 1e-8f) ? 1e-8f : Zv;
        const float rz = __builtin_amdgcn_rcpf(zsafe);     // v_rcp_f32 (~1 ulp)
        const float u = X * rz;
        const float v = Yv * rz;
        const bool valid = (Zv > 0.0f) & (u >= 0.0f) & (u <= (float)(WW - 1)) &
                           (v >= 0.0f) & (v <= (float)(HH - 1));
        // round-to-nearest-even then clip; fmaxf first maps NaN -> 0 (safe index)
        const float uf = fminf(fmaxf(rintf(u), 0.0f), (float)(WW - 1));
        const float vf = fminf(fmaxf(rintf(v), 0.0f), (float)(HH - 1));
        const int ui = (int)uf;
        const int vi = (int)vf;
        const float vval = valid ? 1.0f : 0.0f;

        const float* ip = images + (size_t)img * CCH * HW + (size_t)vi * WW + ui;
        float* op = out + (size_t)img * OCH * NPTS + n;

        // 32-channel gather (stride H*W), NT streaming stores
        #pragma unroll
        for (int ch = 0; ch < CCH; ++ch) {
            const float f = ip[(size_t)ch * HW];
            __builtin_nontemporal_store(f * vval, op + (size_t)ch * NPTS);
        }
        __builtin_nontemporal_store(depth, op + (size_t)32 * NPTS);
        __builtin_nontemporal_store(vval,  op + (size_t)33 * NPTS);

        const float dx = px - s_cam[img][0];
        const float dy = py - s_cam[img][1];
        const float dz = pz - s_cam[img][2];
        // 1/max(sqrt(n2),1e-8) == rsq(max(n2,1e-16)) : single v_rsq_f32
        const float n2  = dx * dx + dy * dy + dz * dz;
        const float inv = __builtin_amdgcn_rsqf(fmaxf(n2, 1e-16f));
        __builtin_nontemporal_store(dx * inv, op + (size_t)34 * NPTS);
        __builtin_nontemporal_store(dy * inv, op + (size_t)35 * NPTS);
        __builtin_nontemporal_store(dz * inv, op + (size_t)36 * NPTS);
    };

    process(ib + 0, d0[0], d0[1], d0[2], d0[3]);
    process(ib + 1, d0[4], d0[5], d0[6], d0[7]);
    process(ib + 4, d1[0], d1[1], d1[2], d1[3]);
    process(ib + 5, d1[4], d1[5], d1[6], d1[7]);
}

extern "C" void kernel_launch(void* const* d_in, const int* in_sizes, int n_in,
                              void* d_out, int out_size, void* d_ws, size_t ws_size,
                              hipStream_t stream) {
    const float* coords = (const float*)d_in[0];   // (3,64,64,64)
    const float* images = (const float*)d_in[1];   // (8,32,480,480)
    const float* trans  = (const float*)d_in[2];   // (8,3,4)
    const float* tcw    = (const float*)d_in[3];   // (8,4,4)
    float* out = (float*)d_out;                    // (8,37,64,64,64)

    // 16384 point-groups of 16, 8 waves (groups) per 256-thread block
    const int blocks = (NPTS / 16) / 8;            // 2048
    proj_gather_wmma_kernel<<<dim3(blocks), dim3(256), 0, stream>>>(
        coords, images, trans, tcw, out);
}